// Taco2DecoderCell_11355893531395
// MI455X (gfx1250) — compile-verified
//
#include <hip/hip_runtime.h>
#include <hip/hip_bf16.h>

typedef __attribute__((ext_vector_type(2))) float v2f;
typedef __attribute__((ext_vector_type(8))) float v8f;
typedef __attribute__((ext_vector_type(4))) unsigned int u32x4;
typedef __attribute__((ext_vector_type(8))) int i32x8;
typedef __attribute__((ext_vector_type(4))) int i32x4;

#define TT 1000
#define DCC 512
#define DAA 128
#define NB 64   // batch

// ---------------------------------------------------------------------------
// Tensor Data Mover 2-D tile load: global (row-major, stride_d0 elements/row)
// -> LDS with per-row padding. fp32 elements. D# layout per CDNA5 ISA §8.
//   pad_icode: row length = (1<<pad_icode)*2 DWORDs between pads
//   pad_acode: pad_acode+1 DWORDs of padding inserted
__device__ __forceinline__ void tdm_load_2d(unsigned lds_addr, const void* gptr,
                                            unsigned tile_d0, unsigned tile_d1,
                                            unsigned tensor_d0, unsigned tensor_d1,
                                            unsigned stride_d0,
                                            unsigned pad_icode, unsigned pad_acode) {
  unsigned long long ga = (unsigned long long)(uintptr_t)gptr;
  u32x4 g0;
  g0[0] = 1u;                                  // count=1, user mode, no gather
  g0[1] = lds_addr;                            // LDS byte address
  g0[2] = (unsigned)(ga & 0xFFFFFFFFu);        // global_addr[31:0]
  g0[3] = (unsigned)((ga >> 32) & 0x01FFFFFFu) // global_addr[56:32]
          | 0x80000000u;                       // type = 2 ("image")
  i32x8 g1;
  g1[0] = (int)((2u << 16)                     // data_size = 4 bytes
                | (1u << 20)                   // pad_enable
                | (pad_icode << 22)
                | (pad_acode << 25));
  g1[1] = (int)((tensor_d0 & 0xFFFFu) << 16);  // dim0[15:0] (atomic_barrier=0)
  g1[2] = (int)(((tensor_d0 >> 16) & 0xFFFFu) | ((tensor_d1 & 0xFFFFu) << 16));
  g1[3] = (int)(((tensor_d1 >> 16) & 0xFFFFu) | ((tile_d0 & 0xFFFFu) << 16));
  g1[4] = (int)(tile_d1 & 0xFFFFu);            // tile_dim1; tile_dim2 = 0
  g1[5] = (int)stride_d0;                      // dim0_stride[31:0]
  g1[6] = 0;                                   // dim0_stride[47:32], dim1_stride lo
  g1[7] = 0;
  i32x4 gz;
  gz[0] = 0; gz[1] = 0; gz[2] = 0; gz[3] = 0;  // 2-D tile: groups 2/3 unused
  i32x8 gz8;
  gz8[0] = 0; gz8[1] = 0; gz8[2] = 0; gz8[3] = 0;
  gz8[4] = 0; gz8[5] = 0; gz8[6] = 0; gz8[7] = 0;
  // 6-arg form (clang-23 / therock headers): extra int32x8 operand, zeroed.
  __builtin_amdgcn_tensor_load_to_lds(g0, g1, gz, gz, gz8, 0);
}

__device__ __forceinline__ unsigned lds_off(const void* p) {
  return (unsigned)(uintptr_t)p;               // low 32 bits = LDS offset
}

// ---------------------------------------------------------------------------
// deterministic "dropout" mask (reference uses JAX threefry(42); bit-exact
// reproduction is impractical here, so use a fixed hash -> deterministic per call)
__device__ __forceinline__ float drop_scale(unsigned seed, unsigned idx) {
  unsigned h = (idx + seed * 0x9E3779B9u);
  h ^= h >> 16; h *= 0x7feb352du;
  h ^= h >> 15; h *= 0x846ca68bu;
  h ^= h >> 16;
  return (h & 1u) ? 2.0f : 0.0f;   // keep prob 0.5, scale 1/(1-p)=2
}

__device__ __forceinline__ float sigmoidf(float x) { return 1.0f / (1.0f + expf(-x)); }

// ---------------------------------------------------------------------------
__global__ void zero_kernel(float* p, int n) {
  int i = blockIdx.x * blockDim.x + threadIdx.x;
  if (i < n) p[i] = 0.0f;
}

// ---------------------------------------------------------------------------
// Fused: mproj[b,t,:] = memory[b,t,:] @ att_wm   (WMMA f32 16x16x4)
//        ctx[b,:]    += w[b,t] * memory[b,t,:]   (atomic f32 accumulation)
// block = 256 threads (8 waves); grid = (8 t-chunks of 128 rows, 64 batches)
// Tiles staged via Tensor Data Mover (async DMA to LDS, row padding in D#).
__global__ void ctx_mproj_kernel(const float* __restrict__ mem,
                                 const float* __restrict__ w,
                                 const float* __restrict__ att_wm,
                                 float* __restrict__ ctx,
                                 float* __restrict__ mproj) {
  __shared__ float sA[128][36];   // memory tile: 128 t-rows x 32 k, pad 32+4
  __shared__ float sB[32][136];   // att_wm tile: 32 k-rows x 128 a, pad 128+8
  __shared__ float sw[128];
  __shared__ float sctx[8][32];

  const int b     = blockIdx.y;
  const int tbase = blockIdx.x * 128;
  const int tid   = threadIdx.x;
  const int lane  = tid & 31;
  const int wave  = tid >> 5;

  if (tid < 128) {
    int t = tbase + tid;
    sw[tid] = (t < TT) ? w[b * TT + t] : 0.0f;
  }

  v8f C[8] = {};
  const float* memB = mem + (size_t)b * TT * DCC;
  const unsigned rows_avail = (unsigned)(TT - tbase);   // TDM zero-fills t >= TT

  for (int kc = 0; kc < 16; ++kc) {          // K = 512 in chunks of 32
    const int k0 = kc * 32;
    __syncthreads();                          // previous tile fully consumed
    if (wave == 0) {
      // memory tile: 32 cols x 128 rows, row stride 512; LDS rows 32+4 dwords
      tdm_load_2d(lds_off(&sA[0][0]), memB + (size_t)tbase * DCC + k0,
                  32u, 128u, 512u, rows_avail, 512u, /*32dw*/4u, /*+4dw*/3u);
      __builtin_amdgcn_s_wait_tensorcnt(0);
    } else if (wave == 1) {
      // att_wm tile: 128 cols x 32 rows, row stride 128; LDS rows 128+8 dwords
      tdm_load_2d(lds_off(&sB[0][0]), att_wm + (size_t)k0 * DAA,
                  128u, 32u, 128u, 32u, 128u, /*128dw*/6u, /*+8dw*/7u);
      __builtin_amdgcn_s_wait_tensorcnt(0);
    }
    __syncthreads();                          // tiles visible to all waves

    // WMMA: each wave owns a 16(t) x 128(a) strip
    {
      const int arow = wave * 16 + (lane & 15);
      const int koff = (lane >> 4) * 2;
      const int bn   = lane & 15;
#pragma unroll
      for (int kk = 0; kk < 32; kk += 4) {
        v2f Af;
        Af.x = sA[arow][kk + koff];
        Af.y = sA[arow][kk + koff + 1];
#pragma unroll
        for (int j = 0; j < 8; ++j) {
          v2f Bf;
          Bf.x = sB[kk + koff][j * 16 + bn];
          Bf.y = sB[kk + koff + 1][j * 16 + bn];
          C[j] = __builtin_amdgcn_wmma_f32_16x16x4_f32(
                     false, Af, false, Bf, (short)0, C[j], false, false);
        }
      }
    }

    // ctx partial sums over this tile's 32 columns
    {
      int col = tid & 31, rg = tid >> 5;
      float s = 0.0f;
#pragma unroll
      for (int rr = 0; rr < 16; ++rr) {
        int row = rg * 16 + rr;
        s += sw[row] * sA[row][col];
      }
      sctx[rg][col] = s;
    }
    __syncthreads();
    if (tid < 32) {
      float acc = 0.0f;
#pragma unroll
      for (int g = 0; g < 8; ++g) acc += sctx[g][tid];
      atomicAdd(&ctx[b * DCC + k0 + tid], acc);
    }
  }

  // store mproj strip
  {
    const int m0 = wave * 16;
#pragma unroll
    for (int r = 0; r < 8; ++r) {
      int t = tbase + m0 + r + ((lane >> 4) * 8);
      if (t < TT) {
#pragma unroll
        for (int j = 0; j < 8; ++j)
          mproj[((size_t)b * TT + t) * DAA + j * 16 + (lane & 15)] = C[j][r];
      }
    }
  }
}

// ---------------------------------------------------------------------------
// PreNet: relu(xf@w1+b1)*drop -> relu(.@w2+b2)*drop. Tiny, one block.
__global__ void prenet_kernel(const float* __restrict__ x,
                              const float* __restrict__ w1, const float* __restrict__ b1,
                              const float* __restrict__ w2, const float* __restrict__ b2,
                              float* __restrict__ xpre) {
  __shared__ float sh[64 * 128];
  int tid = threadIdx.x;
  for (int o = tid; o < 64 * 128; o += 256) {
    int m = o >> 7, j = o & 127;
    float s = b1[j];
    for (int k = 0; k < 80; ++k) s += x[m * 80 + k] * w1[k * 128 + j];
    sh[o] = fmaxf(s, 0.0f) * drop_scale(1u, (unsigned)o);
  }
  __syncthreads();
  for (int o = tid; o < 64 * 128; o += 256) {
    int m = o >> 7, j = o & 127;
    float s = b2[j];
    for (int k = 0; k < 128; ++k) s += sh[(m << 7) + k] * w2[(k << 7) + j];
    xpre[o] = fmaxf(s, 0.0f) * drop_scale(2u, (unsigned)o);
  }
}

// ---------------------------------------------------------------------------
// Generic M=64 GEMM, fp32 WMMA 16x16x4, TDM-staged operands.
// A rows = concat of up to 3 segments (split points s1, s2, all mult of 64).
// W rows = W0 for k < wsplit else W1 (stacked wih;whh). out = A @ W  (no bias)
__global__ void gemm64_kernel(const float* __restrict__ A0,
                              const float* __restrict__ A1,
                              const float* __restrict__ A2,
                              int s1, int s2, int Ktot,
                              const float* __restrict__ W0,
                              const float* __restrict__ W1, int wsplit,
                              int N, float* __restrict__ out) {
  __shared__ float sA[64][68];    // 64 m-rows x 64 k, pad 64+4
  __shared__ float sB[64][136];   // 64 k-rows x 128 n, pad 128+8
  const int tid  = threadIdx.x;
  const int lane = tid & 31;
  const int wave = tid >> 5;
  const int nb0  = blockIdx.x * 128;

  v8f C[4] = {};

  for (int k0 = 0; k0 < Ktot; k0 += 64) {
    __syncthreads();
    if (wave == 0) {
      // A chunk 64(m) x 64(k); segment uniform per chunk (splits mult of 64)
      const float* src;
      unsigned rowlen;
      if (k0 < s1)      { src = A0 + k0;        rowlen = (unsigned)s1; }
      else if (k0 < s2) { src = A1 + (k0 - s1); rowlen = (unsigned)(s2 - s1); }
      else              { src = A2 + (k0 - s2); rowlen = (unsigned)(Ktot - s2); }
      tdm_load_2d(lds_off(&sA[0][0]), src,
                  64u, 64u, 64u, 64u, rowlen, /*64dw*/5u, /*+4dw*/3u);
      __builtin_amdgcn_s_wait_tensorcnt(0);
    } else if (wave == 1) {
      // W chunk 64(k) x 128(n), row stride N
      const float* wsrc = (k0 < wsplit)
                              ? (W0 + (size_t)k0 * N + nb0)
                              : (W1 + (size_t)(k0 - wsplit) * N + nb0);
      tdm_load_2d(lds_off(&sB[0][0]), wsrc,
                  128u, 64u, 128u, 64u, (unsigned)N, /*128dw*/6u, /*+8dw*/7u);
      __builtin_amdgcn_s_wait_tensorcnt(0);
    }
    __syncthreads();

    const int n0l  = wave * 16 + (lane & 15);
    const int koff = (lane >> 4) * 2;
#pragma unroll
    for (int kk = 0; kk < 64; kk += 4) {
      v2f Bf;
      Bf.x = sB[kk + koff][n0l];
      Bf.y = sB[kk + koff + 1][n0l];
#pragma unroll
      for (int mi = 0; mi < 4; ++mi) {
        v2f Af;
        Af.x = sA[mi * 16 + (lane & 15)][kk + koff];
        Af.y = sA[mi * 16 + (lane & 15)][kk + koff + 1];
        C[mi] = __builtin_amdgcn_wmma_f32_16x16x4_f32(
                    false, Af, false, Bf, (short)0, C[mi], false, false);
      }
    }
  }

  const int nc = nb0 + wave * 16 + (lane & 15);
#pragma unroll
  for (int mi = 0; mi < 4; ++mi) {
#pragma unroll
    for (int r = 0; r < 8; ++r) {
      int m = mi * 16 + r + ((lane >> 4) * 8);
      out[(size_t)m * N + nc] = C[mi][r];
    }
  }
}

// ---------------------------------------------------------------------------
// LSTM cell + zoneout (eval-mode mix, PZ = 0.1). gates layout i,f,g,o.
__global__ void lstm_cell_kernel(const float* __restrict__ gates,
                                 const float* __restrict__ bias,
                                 const float* __restrict__ hp,
                                 const float* __restrict__ cp,
                                 float* __restrict__ hn, float* __restrict__ cn) {
  int idx = blockIdx.x * blockDim.x + threadIdx.x;
  if (idx >= 64 * 1024) return;
  int m = idx >> 10, j = idx & 1023;
  const float* g = gates + (size_t)m * 4096;
  float gi = g[j]        + bias[j];
  float gf = g[1024 + j] + bias[1024 + j];
  float gg = g[2048 + j] + bias[2048 + j];
  float go = g[3072 + j] + bias[3072 + j];
  float hprev = hp[idx], cprev = cp[idx];
  float c = sigmoidf(gf) * cprev + sigmoidf(gi) * tanhf(gg);
  float h = sigmoidf(go) * tanhf(c);
  hn[idx] = 0.9f * h + 0.1f * hprev;
  cn[idx] = 0.9f * c + 0.1f * cprev;
}

// ---------------------------------------------------------------------------
// p[b,t] = sigmoid(mask ? v . tanh(q[b]+mproj[b,t]+att_b) : -1e9)
// one wave per (b,t) row; float4 loads + shuffle reduction
__global__ void att_energy_kernel(const float* __restrict__ q,
                                  const float* __restrict__ mproj,
                                  const float* __restrict__ att_b,
                                  const float* __restrict__ att_v,
                                  const unsigned char* __restrict__ mmask,
                                  float* __restrict__ p) {
  int b = blockIdx.y;
  int t = blockIdx.x * 8 + (threadIdx.x >> 5);
  int lane = threadIdx.x & 31;
  if (t >= TT) return;
  int a = lane * 4;
  const float4 mp = *(const float4*)(mproj + ((size_t)b * TT + t) * DAA + a);
  const float4 qq = *(const float4*)(q + b * DAA + a);
  const float4 bb = *(const float4*)(att_b + a);
  const float4 vv = *(const float4*)(att_v + a);
  float s = vv.x * tanhf(qq.x + mp.x + bb.x)
          + vv.y * tanhf(qq.y + mp.y + bb.y)
          + vv.z * tanhf(qq.z + mp.z + bb.z)
          + vv.w * tanhf(qq.w + mp.w + bb.w);
  for (int d = 16; d; d >>= 1) s += __shfl_down(s, d, 32);
  if (lane == 0) {
    float e = mmask[b * TT + t] ? s : -1e9f;
    p[b * TT + t] = sigmoidf(e);
  }
}

// ---------------------------------------------------------------------------
// finalize: x_dec concat, ctx copy, w_new shift-update
__global__ void finalize_kernel(const float* __restrict__ h0n,
                                const float* __restrict__ h1n,
                                const float* __restrict__ ctx,
                                const float* __restrict__ w,
                                const float* __restrict__ p,
                                float* __restrict__ out) {
  int idx = blockIdx.x * 256 + threadIdx.x;
  if (idx < 64 * 2560) {                         // x_dec
    int m = idx / 2560, j = idx % 2560;
    float v = (j < 1024) ? h0n[m * 1024 + j]
            : (j < 2048) ? h1n[m * 1024 + j - 1024] : 0.0f;
    out[idx] = v;
  } else if (idx < 64 * 2560 + 64 * 512) {       // ctx
    int k = idx - 64 * 2560;
    out[idx] = ctx[k];
  } else if (idx < 64 * 2560 + 64 * 512 + 64 * TT) {   // w_new
    int k = idx - (64 * 2560 + 64 * 512);
    int t = k % TT;
    float wn = w[k] * p[k];
    if (t > 0) wn += w[k - 1] * (1.0f - p[k - 1]);
    out[idx] = wn;
  }
}

// ---------------------------------------------------------------------------
extern "C" void kernel_launch(void* const* d_in, const int* in_sizes, int n_in,
                              void* d_out, int out_size, void* d_ws, size_t ws_size,
                              hipStream_t stream) {
  const float* x      = (const float*)d_in[0];
  const float* w      = (const float*)d_in[1];
  const float* h0     = (const float*)d_in[2];
  const float* c0     = (const float*)d_in[3];
  const float* h1     = (const float*)d_in[4];
  const float* c1     = (const float*)d_in[5];
  const float* memory = (const float*)d_in[6];
  const unsigned char* mmask = (const unsigned char*)d_in[7];
  const float* pre_w1 = (const float*)d_in[8];
  const float* pre_b1 = (const float*)d_in[9];
  const float* pre_w2 = (const float*)d_in[10];
  const float* pre_b2 = (const float*)d_in[11];
  const float* wih0   = (const float*)d_in[12];
  const float* whh0   = (const float*)d_in[13];
  const float* b0     = (const float*)d_in[14];
  const float* wih1   = (const float*)d_in[15];
  const float* whh1   = (const float*)d_in[16];
  const float* b1     = (const float*)d_in[17];
  const float* att_wq = (const float*)d_in[18];
  const float* att_wm = (const float*)d_in[19];
  const float* att_b  = (const float*)d_in[20];
  const float* att_v  = (const float*)d_in[21];

  float* ws     = (float*)d_ws;
  float* xpre   = ws;               // 8192
  float* ctx    = ws + 8192;        // 32768
  float* gates0 = ws + 40960;       // 262144
  float* gates1 = ws + 303104;      // 262144
  float* q      = ws + 565248;      // 8192
  float* pbuf   = ws + 573440;      // 64000
  float* mproj  = ws + 637440;      // 8,192,000  (total ~35.3 MB)

  float* out    = (float*)d_out;
  float* o_xdec = out;              // 64*2560
  float* o_h0n  = out + 260608;     // after xdec+ctx+w_new
  float* o_c0n  = out + 326144;
  float* o_h1n  = out + 391680;
  float* o_c1n  = out + 457216;

  zero_kernel<<<128, 256, 0, stream>>>(ctx, 64 * DCC);

  ctx_mproj_kernel<<<dim3(8, NB), 256, 0, stream>>>(memory, w, att_wm, ctx, mproj);

  prenet_kernel<<<1, 256, 0, stream>>>(x, pre_w1, pre_b1, pre_w2, pre_b2, xpre);

  // gates0 = [xpre | ctx | h0] @ [wih0 ; whh0],  K = 128+512+1024 = 1664
  gemm64_kernel<<<32, 256, 0, stream>>>(xpre, ctx, h0, 128, 640, 1664,
                                        wih0, whh0, 640, 4096, gates0);
  lstm_cell_kernel<<<256, 256, 0, stream>>>(gates0, b0, h0, c0, o_h0n, o_c0n);

  // gates1 = [h0n | ctx | h1] @ [wih1 ; whh1],  K = 1024+512+1024 = 2560
  gemm64_kernel<<<32, 256, 0, stream>>>(o_h0n, ctx, h1, 1024, 1536, 2560,
                                        wih1, whh1, 1536, 4096, gates1);
  lstm_cell_kernel<<<256, 256, 0, stream>>>(gates1, b1, h1, c1, o_h1n, o_c1n);

  // q = [h0n | h1n] @ att_wq[0:2048,:]  (ctx slot of x_att is zeros)
  gemm64_kernel<<<1, 256, 0, stream>>>(o_h0n, o_h1n, o_h1n, 1024, 2048, 2048,
                                       att_wq, att_wq, 2048, 128, q);

  att_energy_kernel<<<dim3(125, NB), 256, 0, stream>>>(q, mproj, att_b, att_v,
                                                       mmask, pbuf);

  finalize_kernel<<<1018, 256, 0, stream>>>(o_h0n, o_h1n, ctx, w, pbuf, o_xdec);
}